// DualBranchModel_45775761441089
// MI455X (gfx1250) — compile-verified
//
#include <hip/hip_runtime.h>

#define BATCH 256
#define NTOK  196          // 14*14
#define CH    768
#define NCLS  701
#define NTILE_N 44         // ceil(701/16)

typedef __attribute__((ext_vector_type(2))) float v2f;
typedef __attribute__((ext_vector_type(4))) float v4f;
typedef __attribute__((ext_vector_type(8))) float v8f;

// ---------------------------------------------------------------------------
// Kernel 1: per-token energies M[b,n] = sum_c x[b,n,c].  One wave32 per token.
// Regular-temporal loads: this pass warms L2 (154 MB/branch < 192 MB L2).
// ---------------------------------------------------------------------------
__global__ void energy_kernel(const v4f* __restrict__ x, float* __restrict__ M) {
    int wave = (blockIdx.x * blockDim.x + threadIdx.x) >> 5;   // token in [0, 256*196)
    int lane = threadIdx.x & 31;
    size_t base = (size_t)wave * (CH / 4);
    float s = 0.f;
#pragma unroll
    for (int i = 0; i < 6; ++i) {
        v4f v = x[base + i * 32 + lane];
        s += v.x + v.y + v.z + v.w;
    }
#pragma unroll
    for (int off = 16; off >= 1; off >>= 1)
        s += __shfl_xor(s, off, 32);
    if (lane == 0) M[wave] = s;
}

// ---------------------------------------------------------------------------
// Kernel 2: per-sample SGM selection. One 256-thread block per sample.
// min/max -> normalize -> stable rank (O(N^2), N=196) -> gap diff with
// (sorted_M==0 -> 0) rule -> first-max argmax -> fg/bg weights.
// ---------------------------------------------------------------------------
__global__ void select_kernel(const float* __restrict__ M,
                              float* __restrict__ wfg,
                              float* __restrict__ wbg) {
    __shared__ float sm[NTOK];
    __shared__ float ssorted[NTOK];
    __shared__ float redv[256];
    __shared__ int   redi[256];

    const int tid  = threadIdx.x;
    const int base = blockIdx.x * NTOK;

    float v = (tid < NTOK) ? M[base + tid] : 0.f;

    // ---- min reduction ----
    redv[tid] = (tid < NTOK) ? v : __FLT_MAX__;
    __syncthreads();
    for (int s = 128; s > 0; s >>= 1) {
        if (tid < s) redv[tid] = fminf(redv[tid], redv[tid + s]);
        __syncthreads();
    }
    float mn = redv[0];
    __syncthreads();

    // ---- max reduction ----
    redv[tid] = (tid < NTOK) ? v : -__FLT_MAX__;
    __syncthreads();
    for (int s = 128; s > 0; s >>= 1) {
        if (tid < s) redv[tid] = fmaxf(redv[tid], redv[tid + s]);
        __syncthreads();
    }
    float mx = redv[0];
    __syncthreads();

    float mi = (v - mn) / (mx - mn);
    if (tid < NTOK) sm[tid] = mi;
    __syncthreads();

    // ---- stable rank (matches jnp.argsort ascending) ----
    int rank = 0;
    if (tid < NTOK) {
        for (int j = 0; j < NTOK; ++j) {
            float mj = sm[j];
            rank += (mj < mi) || (mj == mi && j < tid);
        }
        ssorted[rank] = mi;
    }
    __syncthreads();

    // ---- diff + first-max argmax ----
    float dv = -__FLT_MAX__;
    int   di = 0x7FFFFFFF;
    if (tid < NTOK - 1) {
        float s0 = ssorted[tid];
        dv = (s0 == 0.f) ? 0.f : (ssorted[tid + 1] - s0);
        di = tid;
    }
    redv[tid] = dv;
    redi[tid] = di;
    __syncthreads();
    for (int s = 128; s > 0; s >>= 1) {
        if (tid < s) {
            float v2 = redv[tid + s];
            int   i2 = redi[tid + s];
            if (v2 > redv[tid] || (v2 == redv[tid] && i2 < redi[tid])) {
                redv[tid] = v2;
                redi[tid] = i2;
            }
        }
        __syncthreads();
    }
    int k = redi[0];

    if (tid < NTOK) {
        bool  fg   = rank < k;
        float invf = 1.f / (float)((k > 1) ? k : 1);
        float invb = 1.f / (float)(NTOK - k);   // k <= 194, always >= 2
        wfg[base + tid] = fg ? invf : 0.f;
        wbg[base + tid] = fg ? 0.f : invb;
    }
}

// ---------------------------------------------------------------------------
// Kernel 3: weighted sums. One 192-thread block per sample; each thread owns
// one float4 (4 channels). Single pass over features (mostly L2 hits) emits
// both fg and bg rows of the A matrix. Non-temporal (last-use) loads so the
// branch's data is released from L2 before the next branch streams in.
// ---------------------------------------------------------------------------
__global__ void wsum_kernel(const v4f* __restrict__ x,
                            const float* __restrict__ wfg,
                            const float* __restrict__ wbg,
                            v4f* __restrict__ A,      // base of A; branch_row0 selects the branch's 512 rows
                            int branch_row0) {
    __shared__ float wf[NTOK];
    __shared__ float wb[NTOK];
    const int tid = threadIdx.x;   // 0..191
    const int b   = blockIdx.x;    // sample

    for (int i = tid; i < NTOK; i += 192) {
        wf[i] = wfg[b * NTOK + i];
        wb[i] = wbg[b * NTOK + i];
    }
    __syncthreads();

    const v4f* xp = x + (size_t)b * NTOK * (CH / 4);
    v4f aF = {0.f, 0.f, 0.f, 0.f};
    v4f aB = {0.f, 0.f, 0.f, 0.f};
    for (int n = 0; n < NTOK; ++n) {
        v4f vv = __builtin_nontemporal_load(&xp[n * (CH / 4) + tid]);
        float f = wf[n], g = wb[n];
        aF += vv * f;
        aB += vv * g;
    }
    int bgrow = branch_row0 + b;          // bg rows first in each branch's 512
    int fgrow = branch_row0 + 256 + b;
    A[(size_t)bgrow * (CH / 4) + tid] = aB;
    A[(size_t)fgrow * (CH / 4) + tid] = aF;
}

// ---------------------------------------------------------------------------
// Kernel 4: out[1024,701] = A[1024,768] @ W[768,701] + bias, full fp32 via
// V_WMMA_F32_16X16X4_F32. One wave per 16x16 tile; 192 k-steps of 4.
// A fragment: lane l holds A[m0 + l%16, k0 + 2*(l/16) + {0,1}]  (float2)
// B fragment: lane l holds W[k0 + 2*(l/16) + {0,1}, n0 + l%16]
// D layout  : vgpr r -> M = m0 + r + 8*(l/16), N = n0 + l%16
// ---------------------------------------------------------------------------
__global__ void gemm_wmma_kernel(const float* __restrict__ A,
                                 const float* __restrict__ Wm,
                                 const float* __restrict__ bias,
                                 float* __restrict__ out) {
    int wave = (blockIdx.x * blockDim.x + threadIdx.x) >> 5;  // tile id, exact fit
    int lane = threadIdx.x & 31;

    int m0 = (wave / NTILE_N) * 16;
    int n0 = (wave % NTILE_N) * 16;
    int lm = lane & 15;
    int lh = lane >> 4;                 // 0 or 1

    int  row   = m0 + lm;
    int  col   = n0 + lm;
    bool valid = col < NCLS;
    int  colc  = valid ? col : (NCLS - 1);   // clamped, always in-bounds

    const float* arow = A + (size_t)row * CH;
    v8f acc = {0.f, 0.f, 0.f, 0.f, 0.f, 0.f, 0.f, 0.f};

    for (int k0 = 0; k0 < CH; k0 += 4) {
        int kk = k0 + lh * 2;
        v2f a = *(const v2f*)(arow + kk);
        float b0 = Wm[(size_t)kk * NCLS + colc];
        float b1 = Wm[(size_t)(kk + 1) * NCLS + colc];
        v2f bb;
        bb.x = valid ? b0 : 0.f;        // cndmask, no EXEC divergence
        bb.y = valid ? b1 : 0.f;
        acc = __builtin_amdgcn_wmma_f32_16x16x4_f32(
            false, a, false, bb, (short)0, acc, false, false);
    }

    if (valid) {
        float bv = bias[col];
#pragma unroll
        for (int r = 0; r < 8; ++r) {
            int m = m0 + r + (lh ? 8 : 0);
            out[(size_t)m * NCLS + col] = acc[r] + bv;
        }
    }
}

// ---------------------------------------------------------------------------
// Launch: branch-by-branch so each 154 MB branch stays resident in the 192 MB
// L2 between the energy pass and the weighted-sum pass.
// ---------------------------------------------------------------------------
extern "C" void kernel_launch(void* const* d_in, const int* in_sizes, int n_in,
                              void* d_out, int out_size, void* d_ws, size_t ws_size,
                              hipStream_t stream) {
    const float* x1   = (const float*)d_in[0];
    const float* x2   = (const float*)d_in[1];
    const float* Wm   = (const float*)d_in[2];
    const float* bias = (const float*)d_in[3];
    float*       out  = (float*)d_out;
    float*       ws   = (float*)d_ws;

    const int per_branch = BATCH * NTOK;        // 50176 tokens
    float* M   = ws;                            // 2 * 50176
    float* wfg = ws + 2 * per_branch;           // 2 * 50176
    float* wbg = ws + 4 * per_branch;           // 2 * 50176
    float* A   = ws + 6 * per_branch;           // 1024 * 768

    const float* xs[2] = {x1, x2};
    for (int br = 0; br < 2; ++br) {
        const float* x = xs[br];
        float* Mb   = M   + br * per_branch;
        float* wfgb = wfg + br * per_branch;
        float* wbgb = wbg + br * per_branch;

        // 50176 waves, 8 waves/block -> 6272 blocks
        energy_kernel<<<per_branch / 8, 256, 0, stream>>>((const v4f*)x, Mb);
        select_kernel<<<BATCH, 256, 0, stream>>>(Mb, wfgb, wbgb);
        wsum_kernel<<<BATCH, 192, 0, stream>>>((const v4f*)x, wfgb, wbgb,
                                               (v4f*)A, br * 512);
    }

    // 64 x 44 = 2816 tiles, 4 waves/block -> 704 blocks
    gemm_wmma_kernel<<<704, 128, 0, stream>>>(A, Wm, bias, out);
}